// BiDAF_15109694947375
// MI455X (gfx1250) — compile-verified
//
#include <hip/hip_runtime.h>
#include <math.h>
#include <stdint.h>

typedef __attribute__((ext_vector_type(2))) float v2f;
typedef __attribute__((ext_vector_type(8))) float v8f;

#define B_   32
#define T_   1024
#define J_   64
#define D_   256
#define DPAD 260   // padded LDS row stride (floats): lane stride %64 = 4 banks
#define PPAD 68    // padded prob-tile stride
#define CPAD 36    // store-staging stride (16 x 32 tile + pad)
#define WAVES 4    // waves per block, 16 t-rows each -> 64 rows/block

__device__ __forceinline__ v8f wmma_f32(v2f a, v2f b, v8f c) {
  // V_WMMA_F32_16X16X4_F32 : D = A(16x4) x B(4x16) + C(16x16)
  return __builtin_amdgcn_wmma_f32_16x16x4_f32(false, a, false, b, (short)0, c,
                                               false, false);
}

// Async global->LDS B128 copy (GVS mode: SGPR base + 32-bit VGPR offset).
// LDS dest address = low 32 bits of the flat shared pointer (wave-relative).
__device__ __forceinline__ void async_ld_b128(uint32_t lds_addr, uint32_t goff,
                                              const void* sbase) {
  asm volatile("global_load_async_to_lds_b128 %0, %1, %2 offset:0"
               :: "v"(lds_addr), "v"(goff), "s"(sbase)
               : "memory");
}
__device__ __forceinline__ void wait_async0() {
  asm volatile("s_wait_asynccnt 0" ::: "memory");
}

// ---------------------------------------------------------------------------
// Kernel 1: S = (H*w_hu) U^T + h_dot + u_dot + b ; softmax_j ; c2q = A U ;
// writes G chunks 0..2, c2q, and per-(b,t) row-max (for the q2c softmax).
// ---------------------------------------------------------------------------
__global__ void bidaf_main(const float* __restrict__ U,
                           const float* __restrict__ H,
                           const float* __restrict__ w,
                           const float* __restrict__ bptr,
                           float* __restrict__ out,
                           float* __restrict__ Mbuf) {
  extern __shared__ float smem[];
  float* Us      = smem;                        // J_*DPAD
  float* HsAll   = Us + J_ * DPAD;              // WAVES*16*DPAD
  float* PsAll   = HsAll + WAVES * 16 * DPAD;   // WAVES*16*PPAD
  float* CsAll   = PsAll + WAVES * 16 * PPAD;   // WAVES*16*CPAD
  float* w_h_s   = CsAll + WAVES * 16 * CPAD;   // D_
  float* w_hu_s  = w_h_s + D_;                  // D_
  float* u_dot_s = w_hu_s + D_;                 // J_
  float* hd_all  = u_dot_s + J_;                // WAVES*16

  const int tid  = threadIdx.x;
  const int wave = tid >> 5;
  const int lane = tid & 31;
  const int half = lane >> 4;
  const int lm   = lane & 15;

  const int b  = blockIdx.x >> 4;                         // 16 blocks / batch
  const int t0 = (blockIdx.x & 15) * (WAVES * 16) + wave * 16;

  float* Hs   = HsAll + wave * 16 * DPAD;
  float* Ps   = PsAll + wave * 16 * PPAD;
  float* Cs   = CsAll + wave * 16 * CPAD;
  float* hd_s = hd_all + wave * 16;

  // ---- stage w splits (tiny, plain loads) ----
  for (int i = tid; i < D_; i += blockDim.x) {
    w_h_s[i]  = w[i];
    w_hu_s[i] = w[2 * D_ + i];
  }
  // ---- async-stage U[b] (64 x 256) into padded LDS rows ----
  {
    const void* Ug = (const void*)(U + (size_t)b * J_ * D_);
    for (int i = tid; i < J_ * D_ / 4; i += blockDim.x) {
      int j = i >> 6, d4 = i & 63;
      uint32_t lds = (uint32_t)(uintptr_t)&Us[j * DPAD + d4 * 4];
      async_ld_b128(lds, (uint32_t)i * 16u, Ug);
    }
  }
  // ---- async-stage this wave's 16-row H tile ----
  {
    const void* Hg = (const void*)(H + (size_t)(b * T_ + t0) * D_);
    for (int i = lane; i < 16 * 64; i += 32) {
      int row = i >> 6, d4 = i & 63;
      uint32_t lds = (uint32_t)(uintptr_t)&Hs[row * DPAD + d4 * 4];
      async_ld_b128(lds, (uint32_t)(row * 1024 + d4 * 16), Hg);
    }
  }
  wait_async0();
  __syncthreads();

  // ---- u_dot[j] = dot(U[j], w_u) + b  (w_u read via L2) ----
  {
    float bval = bptr[0];
    const float* wu = w + D_;
    for (int j = tid; j < J_; j += blockDim.x) {
      float s = 0.f;
      for (int d = 0; d < D_; ++d) s = fmaf(Us[j * DPAD + d], wu[d], s);
      u_dot_s[j] = s + bval;
    }
  }
  // ---- h_dot per row (wave-local): 2 lanes per row, xor-16 combine ----
  {
    float p = 0.f;
    int dbase = half * 128;
    for (int d = 0; d < 128; ++d)
      p = fmaf(Hs[lm * DPAD + dbase + d], w_h_s[dbase + d], p);
    p += __shfl_xor(p, 16, 32);
    if (lane < 16) hd_s[lm] = p;
  }
  __syncthreads();

  // ================= S = (H*w_hu) * U^T  (16 x 64 per wave) ================
  v8f acc[4];
#pragma unroll
  for (int nt = 0; nt < 4; ++nt)
#pragma unroll
    for (int r = 0; r < 8; ++r) acc[nt][r] = 0.f;

  const int koff = 2 * half;
#pragma unroll 4
  for (int k0 = 0; k0 < D_; k0 += 4) {
    const int ka = k0 + koff;
    float2 hv = *(const float2*)&Hs[lm * DPAD + ka];
    float2 wv = *(const float2*)&w_hu_s[ka];
    v2f a; a.x = hv.x * wv.x; a.y = hv.y * wv.y;
#pragma unroll
    for (int nt = 0; nt < 4; ++nt) {
      float2 uv = *(const float2*)&Us[(nt * 16 + lm) * DPAD + ka];
      v2f bb; bb.x = uv.x; bb.y = uv.y;
      acc[nt] = wmma_f32(a, bb, acc[nt]);
    }
  }

  // ---- add rank-1 terms; row max over j ----
  float rowb[8], rmax[8];
#pragma unroll
  for (int r = 0; r < 8; ++r) { rowb[r] = hd_s[r + 8 * half]; rmax[r] = -INFINITY; }
#pragma unroll
  for (int nt = 0; nt < 4; ++nt) {
    float uj = u_dot_s[nt * 16 + lm];
#pragma unroll
    for (int r = 0; r < 8; ++r) {
      float v = acc[nt][r] + rowb[r] + uj;
      acc[nt][r] = v;
      rmax[r] = fmaxf(rmax[r], v);
    }
  }
#pragma unroll
  for (int r = 0; r < 8; ++r) {
    rmax[r] = fmaxf(rmax[r], __shfl_xor(rmax[r], 1, 32));
    rmax[r] = fmaxf(rmax[r], __shfl_xor(rmax[r], 2, 32));
    rmax[r] = fmaxf(rmax[r], __shfl_xor(rmax[r], 4, 32));
    rmax[r] = fmaxf(rmax[r], __shfl_xor(rmax[r], 8, 32));
  }
  if (lm == 0) {
#pragma unroll
    for (int r = 0; r < 8; ++r)
      Mbuf[(size_t)b * T_ + t0 + 8 * half + r] = rmax[r];
  }

  // ---- softmax over j ----
  float rsum[8];
#pragma unroll
  for (int r = 0; r < 8; ++r) rsum[r] = 0.f;
#pragma unroll
  for (int nt = 0; nt < 4; ++nt)
#pragma unroll
    for (int r = 0; r < 8; ++r) {
      float e = __expf(acc[nt][r] - rmax[r]);
      acc[nt][r] = e;
      rsum[r] += e;
    }
#pragma unroll
  for (int r = 0; r < 8; ++r) {
    rsum[r] += __shfl_xor(rsum[r], 1, 32);
    rsum[r] += __shfl_xor(rsum[r], 2, 32);
    rsum[r] += __shfl_xor(rsum[r], 4, 32);
    rsum[r] += __shfl_xor(rsum[r], 8, 32);
    rsum[r] = 1.f / rsum[r];
  }
  // stash probabilities (C-layout -> LDS, re-read in A-layout)
#pragma unroll
  for (int nt = 0; nt < 4; ++nt)
#pragma unroll
    for (int r = 0; r < 8; ++r)
      Ps[(r + 8 * half) * PPAD + nt * 16 + lm] = acc[nt][r] * rsum[r];
  __syncthreads();

  // ============ c2q = A * U ; stores staged via LDS for b128 writes ========
  float* Gp   = out;
  float* c2qp = out + (size_t)B_ * T_ * 1024;
  const size_t growbase = (size_t)(b * T_ + t0);

  for (int dpair = 0; dpair < 8; ++dpair) {
    const int dbase = dpair * 32;
    // two 16x16 c2q tiles -> Cs[16][32]
#pragma unroll
    for (int sub = 0; sub < 2; ++sub) {
      const int dsub = dbase + sub * 16;
      v8f c2;
#pragma unroll
      for (int r = 0; r < 8; ++r) c2[r] = 0.f;
#pragma unroll 4
      for (int k0 = 0; k0 < J_; k0 += 4) {
        const int ka = k0 + koff;
        float2 pv = *(const float2*)&Ps[lm * PPAD + ka];
        v2f a; a.x = pv.x; a.y = pv.y;
        v2f bb;
        bb.x = Us[ka * DPAD + dsub + lm];
        bb.y = Us[(ka + 1) * DPAD + dsub + lm];
        c2 = wmma_f32(a, bb, c2);
      }
#pragma unroll
      for (int r = 0; r < 8; ++r)
        Cs[(r + 8 * half) * CPAD + sub * 16 + lm] = c2[r];
    }
    // wave-private LDS round-trip (DS ops are in-order within a wave),
    // then b128 stores: 8 lanes cover 128B per output row.
#pragma unroll
    for (int it = 0; it < 4; ++it) {
      const int idx = it * 32 + lane;   // 0..127
      const int row = idx >> 3;         // 16 rows
      const int q   = idx & 7;          // 8 float4 per row
      float4 cv = *(const float4*)&Cs[row * CPAD + q * 4];
      float4 hv = *(const float4*)&Hs[row * DPAD + dbase + q * 4];
      float4 hc;
      hc.x = hv.x * cv.x; hc.y = hv.y * cv.y;
      hc.z = hv.z * cv.z; hc.w = hv.w * cv.w;
      const size_t g = (growbase + row) * 1024 + dbase + q * 4;
      *(float4*)&Gp[g]        = hv;   // G chunk 0: H
      *(float4*)&Gp[g + 256]  = cv;   // G chunk 1: c2q
      *(float4*)&Gp[g + 512]  = hc;   // G chunk 2: H*c2q
      *(float4*)&c2qp[(growbase + row) * 256 + dbase + q * 4] = cv;
    }
  }
}

// ---------------------------------------------------------------------------
// Kernel 2: b_att = softmax_t(max_j S) ; q2c = sum_t b_att * H   (one block/b)
// ---------------------------------------------------------------------------
__global__ void bidaf_q2c(const float* __restrict__ H,
                          const float* __restrict__ Mbuf,
                          float* __restrict__ out) {
  __shared__ float red[256];
  __shared__ float batt[T_];
  const int b = blockIdx.x, tid = threadIdx.x;
  const float* M = Mbuf + (size_t)b * T_;

  float m = -INFINITY;
  for (int t = tid; t < T_; t += 256) m = fmaxf(m, M[t]);
  red[tid] = m; __syncthreads();
  for (int s = 128; s > 0; s >>= 1) {
    if (tid < s) red[tid] = fmaxf(red[tid], red[tid + s]);
    __syncthreads();
  }
  float mx = red[0];
  __syncthreads();

  float ssum = 0.f;
  for (int t = tid; t < T_; t += 256) {
    float e = __expf(M[t] - mx);
    batt[t] = e;
    ssum += e;
  }
  red[tid] = ssum; __syncthreads();
  for (int s = 128; s > 0; s >>= 1) {
    if (tid < s) red[tid] += red[tid + s];
    __syncthreads();
  }
  float inv = 1.f / red[0];
  __syncthreads();

  const float* Hb = H + (size_t)b * T_ * D_;
  float q = 0.f;
  for (int t = 0; t < T_; ++t) q = fmaf(batt[t], Hb[(size_t)t * D_ + tid], q);
  q *= inv;
  out[(size_t)B_ * T_ * 1024 + (size_t)B_ * T_ * 256 + (size_t)b * D_ + tid] = q;
}

// ---------------------------------------------------------------------------
// Kernel 3: G chunk 3 = H * q2c (broadcast over t) — pure streaming pass
// ---------------------------------------------------------------------------
__global__ void bidaf_g4(const float* __restrict__ H, float* __restrict__ out) {
  const float* q2c = out + (size_t)B_ * T_ * 1024 + (size_t)B_ * T_ * 256;
  const int d  = threadIdx.x;
  const int bt = blockIdx.x;       // 0 .. B*T-1
  const int b  = bt >> 10;
  float q = q2c[b * D_ + d];
  float h = H[(size_t)bt * D_ + d];
  out[(size_t)bt * 1024 + 768 + d] = h * q;
}

// ---------------------------------------------------------------------------
extern "C" void kernel_launch(void* const* d_in, const int* in_sizes, int n_in,
                              void* d_out, int out_size, void* d_ws, size_t ws_size,
                              hipStream_t stream) {
  const float* U = (const float*)d_in[0];
  const float* H = (const float*)d_in[1];
  const float* w = (const float*)d_in[2];
  const float* b = (const float*)d_in[3];
  float* out  = (float*)d_out;
  float* Mbuf = (float*)d_ws;   // B*T floats of row-max scratch

  const size_t smem1 =
      (size_t)(J_ * DPAD + WAVES * 16 * DPAD + WAVES * 16 * PPAD +
               WAVES * 16 * CPAD + 2 * D_ + J_ + WAVES * 16) *
      sizeof(float);   // 162,304 B -> 80 LDS blocks -> 2 WGs / WGP

  bidaf_main<<<dim3(B_ * (T_ / (WAVES * 16))), dim3(WAVES * 32), smem1, stream>>>(
      U, H, w, b, out, Mbuf);
  bidaf_q2c<<<dim3(B_), dim3(256), 0, stream>>>(H, Mbuf, out);
  bidaf_g4<<<dim3(B_ * T_), dim3(256), 0, stream>>>(H, out);
}